// DKVMN_57638461112950
// MI455X (gfx1250) — compile-verified
//
#include <hip/hip_runtime.h>
#include <hip/hip_bf16.h>

// ---------------- problem constants ----------------
constexpr int cB  = 64;
constexpr int cL  = 200;
constexpr int cNS = 1000;
constexpr int cD  = 128;
constexpr int cM  = 50;
constexpr int ROWS = cB * cL;          // 12800 flat (b,l) rows

typedef __attribute__((ext_vector_type(2))) float v2f;
typedef __attribute__((ext_vector_type(8))) float v8f;

// masked_r = responses * (responses > -1); x = skills + NS*masked_r
__device__ __forceinline__ int gather_x(const int* skills, const int* resp, int r) {
    int rr = resp[r];
    int mr = (rr > -1) ? rr : 0;
    return skills[r] + cNS * mr;
}

// ---------------------------------------------------------------------------
// Kernel 1: w = softmax(k @ Mk^T)  via WMMA fp32 16x16x4.
// One wave per 16-row l-tile; 4 C-tiles cover cols 0..63 (M=50 padded to 64;
// B addresses clamped to Mk row 49, padded cols masked out of the softmax).
// Row-wise softmax runs directly on the C-fragment layout with shfl_xor
// reductions across the 16 column lanes of each half-wave.
// ---------------------------------------------------------------------------
__global__ __launch_bounds__(256) void k_w(const int* __restrict__ skills,
                                           const float* __restrict__ k_emb,
                                           const float* __restrict__ Mk,
                                           float* __restrict__ w_out) {
    int wave = threadIdx.x >> 5;
    int lane = threadIdx.x & 31;
    int rowtile = blockIdx.x * 8 + wave;       // 800 tiles
    if (rowtile >= ROWS / 16) return;

    int lrow  = lane & 15;
    int khalf = lane >> 4;
    int row   = rowtile * 16 + lrow;
    const float* arow = k_emb + (size_t)skills[row] * cD + khalf * 2;

    v8f acc[4] = {};
    for (int k0 = 0; k0 < cD; k0 += 4) {
        v2f afrag = *(const v2f*)(arow + k0);
#pragma unroll
        for (int t = 0; t < 4; t++) {
            int col  = t * 16 + lrow;
            int ccol = (col < cM) ? col : (cM - 1);   // clamp: no OOB reads of Mk
            v2f bfrag = *(const v2f*)(Mk + (size_t)ccol * cD + k0 + khalf * 2);
            acc[t] = __builtin_amdgcn_wmma_f32_16x16x4_f32(
                false, afrag, false, bfrag, (short)0, acc[t], false, false);
        }
    }

    // masked row-wise softmax over 50 columns
#pragma unroll
    for (int r = 0; r < 8; r++) {
        float mx = -1e30f;
#pragma unroll
        for (int t = 0; t < 4; t++) {
            int col = t * 16 + lrow;
            float v = (col < cM) ? acc[t][r] : -1e30f;
            mx = fmaxf(mx, v);
        }
#pragma unroll
        for (int ofs = 1; ofs < 16; ofs <<= 1) mx = fmaxf(mx, __shfl_xor(mx, ofs, 32));

        float ev[4];
        float s = 0.0f;
#pragma unroll
        for (int t = 0; t < 4; t++) {
            int col = t * 16 + lrow;
            ev[t] = (col < cM) ? __expf(acc[t][r] - mx) : 0.0f;
            s += ev[t];
        }
#pragma unroll
        for (int ofs = 1; ofs < 16; ofs <<= 1) s += __shfl_xor(s, ofs, 32);
        float inv = 1.0f / s;

        int orow = rowtile * 16 + r + khalf * 8;
#pragma unroll
        for (int t = 0; t < 4; t++) {
            int col = t * 16 + lrow;
            if (col < cM) w_out[(size_t)orow * cM + col] = ev[t] * inv;
        }
    }
}

// ---------------------------------------------------------------------------
// Kernel 2: e = sigmoid(v @ e_W^T + e_b), a = tanh(v @ a_W^T + a_b)
// WMMA fp32 16x16x4. One wave = one 16-row tile x one matrix (e or a),
// 8 accumulator tiles cover all 128 output columns. A rows gathered from v_emb.
// ---------------------------------------------------------------------------
__global__ __launch_bounds__(256) void k_ea(const int* __restrict__ skills,
                                            const int* __restrict__ resp,
                                            const float* __restrict__ v_emb,
                                            const float* __restrict__ e_W,
                                            const float* __restrict__ e_b,
                                            const float* __restrict__ a_W,
                                            const float* __restrict__ a_b,
                                            float* __restrict__ e_out,
                                            float* __restrict__ a_out) {
    int wave = threadIdx.x >> 5;
    int lane = threadIdx.x & 31;
    int task = blockIdx.x * 8 + wave;          // 1600 tasks = 800 row-tiles x 2 matrices
    int mat = task & 1;
    int rowtile = task >> 1;
    if (rowtile >= ROWS / 16) return;

    const float* Wm   = mat ? a_W : e_W;
    const float* bias = mat ? a_b : e_b;
    float*       Out  = mat ? a_out : e_out;

    int lrow  = lane & 15;                     // row within tile / col within tile
    int khalf = lane >> 4;                     // K pair select (ISA 16x4 A/B layout)
    int row   = rowtile * 16 + lrow;
    int x     = gather_x(skills, resp, row);
    const float* arow = v_emb + (size_t)x * cD + khalf * 2;

    v8f acc[8] = {};
    for (int k0 = 0; k0 < cD; k0 += 4) {
        v2f afrag = *(const v2f*)(arow + k0);  // A[m][k0+khalf*2 .. +1]
#pragma unroll
        for (int t = 0; t < 8; t++) {
            int col = t * 16 + lrow;
            v2f bfrag = *(const v2f*)(Wm + (size_t)col * cD + k0 + khalf * 2); // B[k][n]=W[n][k]
            acc[t] = __builtin_amdgcn_wmma_f32_16x16x4_f32(
                false, afrag, false, bfrag, (short)0, acc[t], false, false);
        }
    }
    // Epilogue: C layout -> VGPR r holds row r (+8 for lanes>=16), lane%16 = col
#pragma unroll
    for (int t = 0; t < 8; t++) {
        int col = t * 16 + lrow;
        float bv = bias[col];
#pragma unroll
        for (int r = 0; r < 8; r++) {
            int orow = rowtile * 16 + r + khalf * 8;
            float val = acc[t][r] + bv;
            float res = mat ? tanhf(val) : (1.0f / (1.0f + __expf(-val)));
            Out[(size_t)orow * cD + col] = res;
        }
    }
}

// ---------------------------------------------------------------------------
// Kernel 3: sequential scan over L. One block per batch b; Mv lives in LDS.
// read_l = w_l . Mv_pre ; Mv = Mv*(1 - w ox e) + w ox a
// ---------------------------------------------------------------------------
__global__ __launch_bounds__(256) void k_scan(const float* __restrict__ Mv0,
                                              const float* __restrict__ w_ws,
                                              const float* __restrict__ e_ws,
                                              const float* __restrict__ a_ws,
                                              float* __restrict__ read_ws) {
    __shared__ __align__(16) float sMv[cM * cD];
    __shared__ float sw[cM];
    __shared__ float se[cD];
    __shared__ float sa[cD];

    int b = blockIdx.x;
    int t = threadIdx.x;
    for (int i = t; i < cM * cD; i += 256) sMv[i] = Mv0[i];
    __syncthreads();

    for (int l = 0; l < cL; l++) {
        size_t base = (size_t)b * cL + l;
        if (t < cM) sw[t] = w_ws[base * cM + t];
        if (t < cD) se[t] = e_ws[base * cD + t];
        else        sa[t - cD] = a_ws[base * cD + (t - cD)];
        __syncthreads();

        // phase 1: read from PRE-update Mv
        if (t < cD) {
            float rd = 0.0f;
            for (int m = 0; m < cM; m++) rd = fmaf(sw[m], sMv[m * cD + t], rd);
            read_ws[base * cD + t] = rd;
        }
        __syncthreads();

        // phase 2: rank-1 memory update
        for (int i = t; i < cM * cD; i += 256) {
            int m = i >> 7;            // / 128
            int d = i & (cD - 1);
            float wm = sw[m];
            sMv[i] = sMv[i] * (1.0f - wm * se[d]) + wm * sa[d];
        }
        __syncthreads();
    }
}

// ---------------------------------------------------------------------------
// Kernel 4: f = tanh([read,k] @ f_W^T + f_b); p = sigmoid(f . p_W + p_b)
// WMMA fp32, K = 256 (read cols 0..127, gathered k cols 128..255).
// p-dot done with in-register partials + shfl_xor reduction over column lanes.
// ---------------------------------------------------------------------------
__global__ __launch_bounds__(256) void k_f(const int* __restrict__ skills,
                                           const float* __restrict__ k_emb,
                                           const float* __restrict__ read_ws,
                                           const float* __restrict__ f_W,
                                           const float* __restrict__ f_b,
                                           const float* __restrict__ p_W,
                                           const float* __restrict__ p_b,
                                           float* __restrict__ out) {
    int wave = threadIdx.x >> 5;
    int lane = threadIdx.x & 31;
    int rowtile = blockIdx.x * 8 + wave;       // 800 tiles
    if (rowtile >= ROWS / 16) return;

    int lrow  = lane & 15;
    int khalf = lane >> 4;
    int row   = rowtile * 16 + lrow;
    const float* rrow = read_ws + (size_t)row * cD + khalf * 2;
    const float* krow = k_emb + (size_t)skills[row] * cD + khalf * 2;

    v8f acc[8] = {};
    for (int k0 = 0; k0 < 2 * cD; k0 += 4) {
        v2f afrag = (k0 < cD) ? *(const v2f*)(rrow + k0)
                              : *(const v2f*)(krow + (k0 - cD));
#pragma unroll
        for (int t = 0; t < 8; t++) {
            int col = t * 16 + lrow;
            v2f bfrag = *(const v2f*)(f_W + (size_t)col * (2 * cD) + k0 + khalf * 2);
            acc[t] = __builtin_amdgcn_wmma_f32_16x16x4_f32(
                false, afrag, false, bfrag, (short)0, acc[t], false, false);
        }
    }

    float pb = p_b[0];
    float part[8];
#pragma unroll
    for (int r = 0; r < 8; r++) part[r] = 0.0f;
#pragma unroll
    for (int t = 0; t < 8; t++) {
        int col = t * 16 + lrow;
        float fb = f_b[col];
        float pw = p_W[col];
#pragma unroll
        for (int r = 0; r < 8; r++) {
            float fv = tanhf(acc[t][r] + fb);
            part[r] = fmaf(fv, pw, part[r]);
        }
    }
    // reduce the 16 column-lanes within each half-wave (rows differ by +8 across halves)
#pragma unroll
    for (int ofs = 1; ofs < 16; ofs <<= 1) {
#pragma unroll
        for (int r = 0; r < 8; r++) part[r] += __shfl_xor(part[r], ofs, 32);
    }
    if (lrow == 0) {
#pragma unroll
        for (int r = 0; r < 8; r++) {
            int orow = rowtile * 16 + r + khalf * 8;
            int bb = orow / cL;
            int ll = orow % cL;
            float p = 1.0f / (1.0f + __expf(-(part[r] + pb)));
            if (ll > 0) out[(size_t)bb * (cL - 1) + (ll - 1)] = p;
        }
    }
}

// ---------------------------------------------------------------------------
extern "C" void kernel_launch(void* const* d_in, const int* in_sizes, int n_in,
                              void* d_out, int out_size, void* d_ws, size_t ws_size,
                              hipStream_t stream) {
    const int*   skills = (const int*)d_in[0];
    const int*   resp   = (const int*)d_in[1];
    const float* k_emb  = (const float*)d_in[2];
    const float* v_emb  = (const float*)d_in[3];
    const float* Mk     = (const float*)d_in[4];
    const float* Mv0    = (const float*)d_in[5];
    const float* e_W    = (const float*)d_in[6];
    const float* e_b    = (const float*)d_in[7];
    const float* a_W    = (const float*)d_in[8];
    const float* a_b    = (const float*)d_in[9];
    const float* f_W    = (const float*)d_in[10];
    const float* f_b    = (const float*)d_in[11];
    const float* p_W    = (const float*)d_in[12];
    const float* p_b    = (const float*)d_in[13];
    float* out = (float*)d_out;

    // workspace layout (floats): w | e | a | read  (~22.2 MB total)
    float* ws      = (float*)d_ws;
    float* w_ws    = ws;
    float* e_ws    = w_ws + (size_t)ROWS * cM;
    float* a_ws    = e_ws + (size_t)ROWS * cD;
    float* read_ws = a_ws + (size_t)ROWS * cD;

    k_w   <<<(ROWS / 16) / 8, 256, 0, stream>>>(skills, k_emb, Mk, w_ws);
    k_ea  <<<(ROWS / 16) * 2 / 8, 256, 0, stream>>>(skills, resp, v_emb, e_W, e_b, a_W, a_b, e_ws, a_ws);
    k_scan<<<cB, 256, 0, stream>>>(Mv0, w_ws, e_ws, a_ws, read_ws);
    k_f   <<<(ROWS / 16) / 8, 256, 0, stream>>>(skills, k_emb, read_ws, f_W, f_b, p_W, p_b, out);
}